// Low_Rank_RNN_64630667870358
// MI455X (gfx1250) — compile-verified
//
#include <hip/hip_runtime.h>

#define B_    256
#define T_    1024
#define IN_   3
#define HID_  512
#define RANK_ 4
#define OUT_  3

typedef __attribute__((ext_vector_type(2))) float v2f;
typedef __attribute__((ext_vector_type(8))) float v8f;

// tanh(x) = 1 - 2/(e^{2x}+1); clamp keeps e^{2x} finite so rcp never sees inf.
__device__ __forceinline__ float fast_tanh(float x) {
    x = fminf(10.0f, fmaxf(-10.0f, x));
    float e2x = __builtin_amdgcn_exp2f(x * 2.8853900817779268f); // e^{2x} = 2^{2x*log2(e)}
    return 1.0f - 2.0f * __builtin_amdgcn_rcpf(e2x + 1.0f);
}

// ---------------------------------------------------------------------------
// Kernel 1: sequential low-rank recurrence. One wave32 per batch row.
// Lane l owns hidden units j = l*16 .. l*16+15 (all state/weights in VGPRs).
// Per step: rank-4 partial dots -> wave butterfly reduce -> fused preact+tanh
// -> 2KB contiguous wave store of r[b,t,:]. No LDS, no barriers.
// x[b,t+1,:] is software-pipelined one step ahead so its global-load latency
// never lands in the 1024-step serial dependency chain.
// ---------------------------------------------------------------------------
__global__ void __launch_bounds__(32)
rnn_recurrence_kernel(const float* __restrict__ x,
                      const float* __restrict__ hidden,
                      const float* __restrict__ U,
                      const float* __restrict__ V,
                      const float* __restrict__ W_in,
                      const float* __restrict__ b_in,
                      float* __restrict__ h_last,
                      float* __restrict__ r_out) {
    const int b    = blockIdx.x;        // 256 blocks, one wave each
    const int lane = threadIdx.x;       // 0..31 (wave32)
    const int j0   = lane * 16;

    float h[16], Vw[4][16], Uw[4][16], Wi[3][16], bi[16];
#pragma unroll
    for (int k = 0; k < 16; ++k) {
        const int j = j0 + k;
        h[k]  = hidden[b * HID_ + j];
        bi[k] = b_in[j];
#pragma unroll
        for (int r = 0; r < RANK_; ++r) {
            Vw[r][k] = V[r * HID_ + j];
            Uw[r][k] = U[r * HID_ + j];
        }
#pragma unroll
        for (int i = 0; i < IN_; ++i) Wi[i][k] = W_in[j * IN_ + i];
    }

    const float* xb = x + (size_t)b * T_ * IN_;
    float*       rb = r_out + (size_t)b * T_ * HID_;

    // prologue of the x pipeline
    float x0 = xb[0], x1 = xb[1], x2 = xb[2];

    for (int t = 0; t < T_; ++t) {
        // ---- issue next step's x load now; consumed at the bottom ----
        const int tn = (t + 1 < T_) ? (t + 1) : (T_ - 1);
        const float nx0 = xb[tn * IN_ + 0];
        const float nx1 = xb[tn * IN_ + 1];
        const float nx2 = xb[tn * IN_ + 2];

        // p[r] = sum_j h[j] * V[r][j] : per-lane partials
        float a0 = 0.f, a1 = 0.f, a2 = 0.f, a3 = 0.f;
#pragma unroll
        for (int k = 0; k < 16; ++k) {
            const float hv = h[k];
            a0 = fmaf(hv, Vw[0][k], a0);
            a1 = fmaf(hv, Vw[1][k], a1);
            a2 = fmaf(hv, Vw[2][k], a2);
            a3 = fmaf(hv, Vw[3][k], a3);
        }
        // wave32 butterfly reduce; leaves full sums in every lane
#pragma unroll
        for (int m = 16; m > 0; m >>= 1) {
            a0 += __shfl_xor(a0, m, 32);
            a1 += __shfl_xor(a1, m, 32);
            a2 += __shfl_xor(a2, m, 32);
            a3 += __shfl_xor(a3, m, 32);
        }

#pragma unroll
        for (int k = 0; k < 16; ++k) {
            float pre = bi[k];
            pre = fmaf(x0, Wi[0][k], pre);
            pre = fmaf(x1, Wi[1][k], pre);
            pre = fmaf(x2, Wi[2][k], pre);
            pre = fmaf(a0, Uw[0][k], pre);
            pre = fmaf(a1, Uw[1][k], pre);
            pre = fmaf(a2, Uw[2][k], pre);
            pre = fmaf(a3, Uw[3][k], pre);
            h[k] = fast_tanh(pre);
        }

        // wave writes one contiguous 2KB row: r[b,t,:]
        float4* dst = (float4*)(rb + (size_t)t * HID_ + j0);
#pragma unroll
        for (int q = 0; q < 4; ++q)
            dst[q] = make_float4(h[4 * q + 0], h[4 * q + 1], h[4 * q + 2], h[4 * q + 3]);

        // rotate the x pipeline (load latency was hidden behind this step)
        x0 = nx0; x1 = nx1; x2 = nx2;
    }

#pragma unroll
    for (int k = 0; k < 16; ++k) h_last[b * HID_ + j0 + k] = h[k];
}

// ---------------------------------------------------------------------------
// Kernel 2: output projection out[M,3] = r[M,512] @ W_out^T + b_out, M = B*T.
// Full f32 via v_wmma_f32_16x16x4_f32 (exact). One wave per 16-row tile,
// 128 accumulating WMMAs over K=512. N padded 3 -> 16 with a branchless mask
// so EXEC is all-1s for every WMMA. Unroll 8 => 16 loads in flight/clause.
// ---------------------------------------------------------------------------
__global__ void __launch_bounds__(256)
out_proj_wmma_kernel(const float* __restrict__ r,
                     const float* __restrict__ W_out,
                     const float* __restrict__ b_out,
                     float* __restrict__ out) {
    const int wave = blockIdx.x * (blockDim.x >> 5) + (threadIdx.x >> 5);
    const int lane = threadIdx.x & 31;
    const int row0 = wave * 16;            // tile of 16 rows of M = B*T
    const int m    = lane & 15;            // A-layout: M index for this lane
    const int koff = (lane >> 4) * 2;      // lanes 16-31 hold K=2,3

    const float* arow = r + (size_t)(row0 + m) * HID_ + koff;
    // B operand: W_out^T zero-padded to 16 cols; clamp row, mask value.
    const int    nclmp = (m < OUT_) ? m : (OUT_ - 1);
    const float  bmask = (m < OUT_) ? 1.0f : 0.0f;
    const float* wrow  = W_out + (size_t)nclmp * HID_ + koff;

    v8f c = {0.f, 0.f, 0.f, 0.f, 0.f, 0.f, 0.f, 0.f};

#pragma unroll 8
    for (int cb = 0; cb < HID_ / 4; ++cb) {
        const int j = cb * 4;
        v2f a, bb;
        a.x  = arow[j];
        a.y  = arow[j + 1];
        bb.x = wrow[j] * bmask;
        bb.y = wrow[j + 1] * bmask;
        // D = A(16x4) * B(4x16) + C  -- v_wmma_f32_16x16x4_f32
        c = __builtin_amdgcn_wmma_f32_16x16x4_f32(
                /*neg_a=*/false, a, /*neg_b=*/false, bb,
                /*c_mod=*/(short)0, c, /*reuse_a=*/false, /*reuse_b=*/false);
    }

    // C/D layout: VGPR v -> M=v (lanes 0-15, N=lane) / M=v+8 (lanes 16-31, N=lane-16)
    const int n     = (lane < 16) ? lane : (lane - 16);
    const int mbase = row0 + ((lane < 16) ? 0 : 8);
    if (n < OUT_) {
        const float bo = b_out[n];
#pragma unroll
        for (int v = 0; v < 8; ++v)
            out[(size_t)(mbase + v) * OUT_ + n] = c[v] + bo;
    }
}

extern "C" void kernel_launch(void* const* d_in, const int* in_sizes, int n_in,
                              void* d_out, int out_size, void* d_ws, size_t ws_size,
                              hipStream_t stream) {
    const float* x      = (const float*)d_in[0];
    const float* hidden = (const float*)d_in[1];
    const float* U      = (const float*)d_in[2];
    const float* V      = (const float*)d_in[3];
    const float* W_in   = (const float*)d_in[4];
    const float* b_in   = (const float*)d_in[5];
    const float* W_out  = (const float*)d_in[6];
    const float* b_out  = (const float*)d_in[7];

    float* out    = (float*)d_out;                       // [B,T,OUT]
    float* h_last = out + (size_t)B_ * T_ * OUT_;        // [B,HID]
    float* r      = h_last + (size_t)B_ * HID_;          // [B,T,HID]

    rnn_recurrence_kernel<<<B_, 32, 0, stream>>>(x, hidden, U, V, W_in, b_in, h_last, r);

    const int waves  = (B_ * T_) / 16;                   // 16384 tiles
    const int blocks = waves / 8;                        // 8 waves / 256-thread block
    out_proj_wmma_kernel<<<blocks, 256, 0, stream>>>(r, W_out, b_out, out);
}